// QRNN_Paper_8632884265189
// MI455X (gfx1250) — compile-verified
//
#include <hip/hip_runtime.h>

typedef float v2f __attribute__((ext_vector_type(2)));
typedef float v4f __attribute__((ext_vector_type(4)));
typedef float v8f __attribute__((ext_vector_type(8)));

// Problem: B=256, T=32, n=16 qubits, n_data=8.
// State index s (16 bits): bit (15-j) <-> qubit j  (qubit 0 = MSB), matching
// the reference's row-major (2,)*16 axis order.
#define NSTATE 65536
#define NBATCH 256
#define NT     32

// ws layout (float offsets)
#define OFF_DRE   0u        // 65536: Re of RZZ-ring diagonal
#define OFF_DIM   65536u    // 65536: Im of diagonal
#define OFF_GRE   131072u   // 64: fixed per-qubit gate G_j = RX*RZ*RX (2x2, Re)
#define OFF_GIM   131136u   // 64: Im
#define OFF_UAR   131200u   // 256: group A (qubits 12..15, state bits 0-3) Re
#define OFF_UAI   131456u   // 256: Im
#define OFF_UAN   131712u   // 256: -Im
#define OFF_UBR   131968u   // group B (qubits 8..11, state bits 4-7)
#define OFF_UBI   132224u
#define OFF_UBN   132480u
#define OFF_SLABS 262144u   // state slabs start here (1 MB header)
#define SLAB_F    131072u   // per-slab floats: 65536 Re + 65536 Im

// ---------------- WMMA helpers (fp32 matrix pipe) ----------------
__device__ __forceinline__ v8f wmma4(v2f a, v2f b, v8f c) {
  // D(16x16 f32) = A(16x4) * B(4x16) + C   — chained 4x along K for K=16.
  return __builtin_amdgcn_wmma_f32_16x16x4_f32(
      /*neg_a=*/false, a, /*neg_b=*/false, b,
      /*c_mod=*/(short)0, c, /*reuse_a=*/false, /*reuse_b=*/false);
}

// B fragment for one 16x16 matrix U used as B = U^T (B[k][n] = U[n][k]).
// Assumed 4x16 f32 B layout (mirrors documented 16x4 A layout):
//   VGPR0: lanes0-15 -> row k=4kk+0, lanes16-31 -> row k=4kk+2
//   VGPR1: rows k=4kk+1 / k=4kk+3
struct BFrag { v2f r[4], i[4], n[4]; };

__device__ __forceinline__ BFrag load_bfrag(const float* Ur, const float* Ui,
                                            const float* Un, int nn, int kh) {
  BFrag f;
#pragma unroll
  for (int kk = 0; kk < 4; ++kk) {
    const int k0 = 4 * kk + 2 * kh;
    f.r[kk] = *(const v2f*)(Ur + nn * 16 + k0);
    f.i[kk] = *(const v2f*)(Ui + nn * 16 + k0);
    f.n[kk] = *(const v2f*)(Un + nn * 16 + k0);
  }
  return f;
}

// Complex (16xK)*(Kx16): Re = Ar*Ur^T + Ai*(-Ui)^T ; Im = Ar*Ui^T + Ai*Ur^T
__device__ __forceinline__ void cmatmul(const v2f* are, const v2f* aim,
                                        const BFrag& B, v8f& fre, v8f& fim) {
#pragma unroll
  for (int kk = 0; kk < 4; ++kk) {
    fre = wmma4(are[kk], B.r[kk], fre);
    fre = wmma4(aim[kk], B.n[kk], fre);
    fim = wmma4(are[kk], B.i[kk], fim);
    fim = wmma4(aim[kk], B.r[kk], fim);
  }
}

// One fused pass: apply two 4-qubit group gates (B1 on the K axis, then B2 on
// the other axis via an LDS transpose). STRIDED=false: bits 0-7 tiles
// (contiguous 256 amps). STRIDED=true: bits 8-15 tiles (stride-256 gather),
// with the RZZ diagonal fused into writeback.
template <bool STRIDED>
__device__ __forceinline__ void apply_pass(float* __restrict__ sre,
                                           float* __restrict__ sim,
                                           const BFrag& B1, const BFrag& B2,
                                           float* Tre, float* Tim,
                                           const float* __restrict__ dre,
                                           const float* __restrict__ dim_,
                                           int nn, int kh, int wv) {
  for (int tile = wv; tile < 256; tile += 8) {
    v2f are[4], aim[4];
    if (!STRIDED) {
      const int tb = tile << 8;
#pragma unroll
      for (int kk = 0; kk < 4; ++kk) {
        const int k0 = 4 * kk + 2 * kh;
        are[kk] = *(const v2f*)(sre + tb + nn * 16 + k0);
        aim[kk] = *(const v2f*)(sim + tb + nn * 16 + k0);
      }
      if (tile + 8 < 256)  // hint next tile into cache (global_prefetch)
        __builtin_prefetch(sre + ((tile + 8) << 8) + nn * 16, 0, 1);
    } else {
#pragma unroll
      for (int kk = 0; kk < 4; ++kk) {
        const int k0 = 4 * kk + 2 * kh;
        const int s0 = ((nn * 16 + k0) << 8) + tile;
        v2f tr, ti;
        tr.x = sre[s0]; tr.y = sre[s0 + 256];
        ti.x = sim[s0]; ti.y = sim[s0 + 256];
        are[kk] = tr; aim[kk] = ti;
      }
    }

    v8f fre = {}, fim = {};
    cmatmul(are, aim, B1, fre, fim);

    // Transpose 16x16 fragment through per-wave LDS scratch (row stride 20
    // keeps 8/16B alignment; same-wave DS ops are in-order, no barrier).
#pragma unroll
    for (int v = 0; v < 8; ++v) {
      const int mm = v + 8 * kh;   // D layout: VGPR v -> M = v (+8 for hi half)
      Tre[nn * 20 + mm] = fre[v];  // N = lane&15
      Tim[nn * 20 + mm] = fim[v];
    }
    v2f cre[4], cim[4];
#pragma unroll
    for (int kk = 0; kk < 4; ++kk) {
      const int k0 = 4 * kk + 2 * kh;
      cre[kk] = *(const v2f*)(Tre + nn * 20 + k0);
      cim[kk] = *(const v2f*)(Tim + nn * 20 + k0);
    }

    v8f gre = {}, gim = {};
    cmatmul(cre, cim, B2, gre, gim);

    if (!STRIDED) {
      const int tb = tile << 8;
      float* pr = sre + tb + nn * 16 + 8 * kh;  // Final[r=nn][c=v+8kh]
      float* pi = sim + tb + nn * 16 + 8 * kh;
      v4f q0 = {gre[0], gre[1], gre[2], gre[3]};
      v4f q1 = {gre[4], gre[5], gre[6], gre[7]};
      v4f q2 = {gim[0], gim[1], gim[2], gim[3]};
      v4f q3 = {gim[4], gim[5], gim[6], gim[7]};
      *(v4f*)pr = q0; *(v4f*)(pr + 4) = q1;
      *(v4f*)pi = q2; *(v4f*)(pi + 4) = q3;
    } else {
#pragma unroll
      for (int v = 0; v < 8; ++v) {
        const int mm = v + 8 * kh;
        const int s = ((nn * 16 + mm) << 8) + tile;
        const float dr = dre[s], di = dim_[s];  // fused RZZ diagonal
        const float fr = gre[v], fi = gim[v];
        sre[s] = fr * dr - fi * di;
        sim[s] = fr * di + fi * dr;
      }
    }
  }
}

// ---------------- setup kernels ----------------
__device__ __forceinline__ void mm2(float* Cr, float* Ci, const float* Ar,
                                    const float* Ai, const float* Br,
                                    const float* Bi) {
#pragma unroll
  for (int r = 0; r < 2; ++r)
#pragma unroll
    for (int c = 0; c < 2; ++c) {
      float xr = 0.f, xi = 0.f;
#pragma unroll
      for (int k = 0; k < 2; ++k) {
        xr += Ar[r * 2 + k] * Br[k * 2 + c] - Ai[r * 2 + k] * Bi[k * 2 + c];
        xi += Ar[r * 2 + k] * Bi[k * 2 + c] + Ai[r * 2 + k] * Br[k * 2 + c];
      }
      Cr[r * 2 + c] = xr; Ci[r * 2 + c] = xi;
    }
}

__global__ void k_setup_gates(const float* __restrict__ w, float* __restrict__ ws) {
  int j = threadIdx.x;
  if (j >= 16) return;
  float h0 = 0.5f * w[j * 4 + 0], h1 = 0.5f * w[j * 4 + 1], h2 = 0.5f * w[j * 4 + 2];
  float c0 = cosf(h0), s0 = sinf(h0);
  float c1 = cosf(h1), s1 = sinf(h1);
  float c2v = cosf(h2), s2 = sinf(h2);
  float Xr[4] = {c0, 0.f, 0.f, c0},  Xi[4] = {0.f, -s0, -s0, 0.f};   // RX(a0)
  float Zr[4] = {c1, 0.f, 0.f, c1},  Zi[4] = {-s1, 0.f, 0.f, s1};    // RZ(a1)
  float Yr[4] = {c2v, 0.f, 0.f, c2v}, Yi[4] = {0.f, -s2, -s2, 0.f};  // RX(a2)
  float Tr[4], Ti[4], Gr[4], Gi[4];
  mm2(Tr, Ti, Zr, Zi, Xr, Xi);   // RZ*RX
  mm2(Gr, Gi, Yr, Yi, Tr, Ti);   // RX*RZ*RX
#pragma unroll
  for (int e = 0; e < 4; ++e) {
    ws[OFF_GRE + j * 4 + e] = Gr[e];
    ws[OFF_GIM + j * 4 + e] = Gi[e];
  }
}

__global__ void k_setup_groups(float* __restrict__ ws) {
  int tid = threadIdx.x;
  int r = tid >> 4, c = tid & 15;
  float ar = 1.f, ai = 0.f, br = 1.f, bi = 0.f;
#pragma unroll
  for (int i = 0; i < 4; ++i) {
    int rb = (r >> (3 - i)) & 1, cb = (c >> (3 - i)) & 1;
    int e = rb * 2 + cb;
    float gr = ws[OFF_GRE + (12 + i) * 4 + e], gi = ws[OFF_GIM + (12 + i) * 4 + e];
    float nr = ar * gr - ai * gi, ni = ar * gi + ai * gr; ar = nr; ai = ni;
    gr = ws[OFF_GRE + (8 + i) * 4 + e]; gi = ws[OFF_GIM + (8 + i) * 4 + e];
    nr = br * gr - bi * gi; ni = br * gi + bi * gr; br = nr; bi = ni;
  }
  ws[OFF_UAR + tid] = ar; ws[OFF_UAI + tid] = ai; ws[OFF_UAN + tid] = -ai;
  ws[OFF_UBR + tid] = br; ws[OFF_UBI + tid] = bi; ws[OFF_UBN + tid] = -bi;
}

__global__ void k_setup_diag(const float* __restrict__ w, float* __restrict__ ws) {
  int s = blockIdx.x * 256 + threadIdx.x;
  float ang = 0.f;
#pragma unroll
  for (int j = 0; j < 15; ++j) {  // edge (j, j+1) uses weights[j+1, 3]
    int bj = (s >> (15 - j)) & 1, bj1 = (s >> (14 - j)) & 1;
    float th = w[(j + 1) * 4 + 3];
    ang += (bj == bj1) ? th : -th;
  }
  {  // edge (15, 0) uses weights[0, 3]
    int b15 = s & 1, b0 = (s >> 15) & 1;
    float th = w[3];
    ang += (b15 == b0) ? th : -th;
  }
  float h = 0.5f * ang;            // phase = exp(-i*ang/2)
  ws[OFF_DRE + s] = cosf(h);
  ws[OFF_DIM + s] = -sinf(h);
}

__global__ void k_init(float* __restrict__ ws) {
  float* slab = ws + OFF_SLABS + (size_t)blockIdx.x * SLAB_F;
  int tid = threadIdx.x;
  v4f z = {};
  for (int i = tid; i < (int)(SLAB_F / 4); i += 256) ((v4f*)slab)[i] = z;
  if (tid == 0) slab[0] = 1.0f;  // |00...0>
}

// ---------------- main kernel: one workgroup per batch element ----------------
__global__ __launch_bounds__(256) void k_qrnn_main(
    const float* __restrict__ x, const float* __restrict__ wout,
    const float* __restrict__ bout, float* __restrict__ ws, int bBase,
    float* __restrict__ out) {
  __shared__ __align__(16) float sGre[64], sGim[64];
  __shared__ __align__(16) float sMre[32], sMim[32];
  __shared__ __align__(16) float sUCr[256], sUCi[256], sUCn[256];
  __shared__ __align__(16) float sUDr[256], sUDi[256], sUDn[256];
  __shared__ __align__(16) float sT[8][2][320];  // per-wave 16x20 Re/Im scratch
  __shared__ float sCoef[256];
  __shared__ float sRed[256];

  const int tid = threadIdx.x;
  const int nn = tid & 15;          // fragment N / A-row index
  const int kh = (tid >> 4) & 1;    // lane half
  const int wv = tid >> 5;          // wave id (wave32)
  const int b = bBase + blockIdx.x;

  float* sre = ws + OFF_SLABS + (size_t)blockIdx.x * SLAB_F;
  float* sim = sre + NSTATE;
  const float* dre = ws + OFF_DRE;
  const float* dim_ = ws + OFF_DIM;

  if (tid < 64) { sGre[tid] = ws[OFF_GRE + tid]; sGim[tid] = ws[OFF_GIM + tid]; }

  // Fixed-group B fragments, loaded once (loop invariant).
  BFrag BA = load_bfrag(ws + OFF_UAR, ws + OFF_UAI, ws + OFF_UAN, nn, kh);
  BFrag BB = load_bfrag(ws + OFF_UBR, ws + OFF_UBI, ws + OFF_UBN, nn, kh);
  __syncthreads();

  for (int t = 0; t < NT; ++t) {
    // Ry(arccos(x)) via half-angle identities (theta in [0,pi]).
    float xt = x[b * NT + t];
    xt = fminf(1.f, fmaxf(-1.f, xt));
    float cth = sqrtf(0.5f * (1.f + xt));
    float sth = sqrtf(0.5f * (1.f - xt));

    // M_j = G_j * Ry  for data qubits j=0..7 (32 threads, one entry each).
    if (tid < 32) {
      int j = tid >> 2, e = tid & 3, rr = e >> 1, cc = e & 1;
      float ry0 = cc ? -sth : cth;
      float ry1 = cc ? cth : sth;
      float g0r = sGre[j * 4 + rr * 2 + 0], g1r = sGre[j * 4 + rr * 2 + 1];
      float g0i = sGim[j * 4 + rr * 2 + 0], g1i = sGim[j * 4 + rr * 2 + 1];
      sMre[tid] = g0r * ry0 + g1r * ry1;
      sMim[tid] = g0i * ry0 + g1i * ry1;
    }
    __syncthreads();

    // Kron-build group C (qubits 4..7, bits 8-11) and D (qubits 0..3, bits 12-15).
    {
      int r = tid >> 4, c = tid & 15;
      float crv = 1.f, civ = 0.f, drv = 1.f, div = 0.f;
#pragma unroll
      for (int i = 0; i < 4; ++i) {
        int rb = (r >> (3 - i)) & 1, cb = (c >> (3 - i)) & 1;
        int e = rb * 2 + cb;
        float mr = sMre[(4 + i) * 4 + e], mi = sMim[(4 + i) * 4 + e];
        float nr = crv * mr - civ * mi, ni = crv * mi + civ * mr; crv = nr; civ = ni;
        mr = sMre[i * 4 + e]; mi = sMim[i * 4 + e];
        nr = drv * mr - div * mi; ni = drv * mi + div * mr; drv = nr; div = ni;
      }
      sUCr[tid] = crv; sUCi[tid] = civ; sUCn[tid] = -civ;
      sUDr[tid] = drv; sUDi[tid] = div; sUDn[tid] = -div;
    }
    __syncthreads();

    // Pass 1: bits 0-7 (qubits 8..15, fixed gates).
    apply_pass<false>(sre, sim, BA, BB, &sT[wv][0][0], &sT[wv][1][0], dre, dim_,
                      nn, kh, wv);
    __syncthreads();

    // Pass 2: bits 8-15 (Ry-bearing groups) + fused RZZ diagonal.
    BFrag BC = load_bfrag(sUCr, sUCi, sUCn, nn, kh);
    BFrag BD = load_bfrag(sUDr, sUDi, sUDn, nn, kh);
    apply_pass<true>(sre, sim, BC, BD, &sT[wv][0][0], &sT[wv][1][0], dre, dim_,
                     nn, kh, wv);
    __syncthreads();
  }

  // <Z_j> for j=0..7 folded into one weighted sum: coef depends only on s>>8.
  {
    float cf = 0.f;
#pragma unroll
    for (int j = 0; j < 8; ++j)
      cf += wout[j] * (1.f - 2.f * (float)((tid >> (7 - j)) & 1));
    sCoef[tid] = cf;
  }
  __syncthreads();
  float acc = 0.f;
  for (int s = tid; s < NSTATE; s += 256) {
    float re = sre[s], im = sim[s];
    acc += (re * re + im * im) * sCoef[s >> 8];
  }
  sRed[tid] = acc;
  __syncthreads();
  for (int off = 128; off > 0; off >>= 1) {
    if (tid < off) sRed[tid] += sRed[tid + off];
    __syncthreads();
  }
  if (tid == 0) out[b] = sRed[0] + bout[0];
}

// ---------------- host ----------------
extern "C" void kernel_launch(void* const* d_in, const int* in_sizes, int n_in,
                              void* d_out, int out_size, void* d_ws,
                              size_t ws_size, hipStream_t stream) {
  (void)in_sizes; (void)n_in; (void)out_size;
  const float* x = (const float*)d_in[0];       // [256,32]
  const float* w = (const float*)d_in[1];       // [16,4]
  const float* wout = (const float*)d_in[2];    // [1,8]
  const float* bout = (const float*)d_in[3];    // [1]
  float* out = (float*)d_out;                   // [256,1]
  float* ws = (float*)d_ws;

  k_setup_gates<<<1, 16, 0, stream>>>(w, ws);
  k_setup_groups<<<1, 256, 0, stream>>>(ws);
  k_setup_diag<<<256, 256, 0, stream>>>(w, ws);

  // Fit as many 512KB state slabs as the workspace allows; chunk the batch.
  size_t wsF = ws_size / sizeof(float);
  long avail = (long)wsF - (long)OFF_SLABS;
  int slabs = 1;
  if (avail >= (long)SLAB_F) {
    long nsl = avail / (long)SLAB_F;
    slabs = (nsl > 256) ? 256 : (int)nsl;
  }
  for (int b0 = 0; b0 < NBATCH; b0 += slabs) {
    int nb = NBATCH - b0; if (nb > slabs) nb = slabs;
    k_init<<<nb, 256, 0, stream>>>(ws);
    k_qrnn_main<<<nb, 256, 0, stream>>>(x, wout, bout, ws, b0, out);
  }
}